// UncertaintyEstimator_31885837205808
// MI455X (gfx1250) — compile-verified
//
#include <hip/hip_runtime.h>
#include <hip/hip_bf16.h>

typedef float v2f __attribute__((ext_vector_type(2)));
typedef float v8f __attribute__((ext_vector_type(8)));
typedef __attribute__((__vector_size__(4 * sizeof(int)))) int vi4;     // 16B payload type

#define EMBED_DIM 1024
#define HIDDEN    128
#define N_QUERY   65536
#define N_PROTO   512

#define M_TILE     64
#define K_CHUNK    128
#define LDS_STRIDE 132   // 128 + 4 pad -> bank step 4 per row, conflict-free b64 frag reads

// ---- gfx1250 async global->LDS path (guarded; falls back to sync staging) ----
#if __has_builtin(__builtin_amdgcn_global_load_async_to_lds_b128)
#define HAVE_ASYNC_LDS 1
#else
#define HAVE_ASYNC_LDS 0
#endif

#if __has_builtin(__builtin_amdgcn_s_wait_asynccnt)
#define WAIT_ASYNC0() __builtin_amdgcn_s_wait_asynccnt(0)
#else
#define WAIT_ASYNC0() asm volatile("s_wait_asynccnt 0x0" ::: "memory")
#endif

#if HAVE_ASYNC_LDS
typedef __attribute__((address_space(1))) vi4* gptr_vi4;   // global int4*
typedef __attribute__((address_space(3))) vi4* lptr_vi4;   // LDS int4*
__device__ __forceinline__ void async_ld128(const float* g, float* l) {
    __builtin_amdgcn_global_load_async_to_lds_b128(
        (gptr_vi4)(uintptr_t)g,
        (lptr_vi4)(uint32_t)(uintptr_t)l,   // low 32 bits of generic LDS ptr = LDS offset
        0, 0);
}
#endif

// ---------------------------------------------------------------------------
// Kernel 0: per-row argmin over 512 distances. One wave32 per row, float4 scan.
// ---------------------------------------------------------------------------
__global__ __launch_bounds__(256) void argmin_rows(const float* __restrict__ dist,
                                                   int* __restrict__ idx_out) {
    const int row  = blockIdx.x * 8 + (threadIdx.x >> 5);
    const int lane = threadIdx.x & 31;
    const float4* __restrict__ d4 = (const float4*)(dist + (size_t)row * N_PROTO);

    float best = 0.f;
    int   bi   = 0;
    #pragma unroll
    for (int i = 0; i < N_PROTO / 128; ++i) {          // 4 float4 loads per lane
        const float4 v   = d4[lane + 32 * i];
        const int    bse = 4 * lane + 128 * i;          // in-lane indices strictly increase
        if (i == 0)          { best = v.x; bi = bse;     }
        else if (v.x < best) { best = v.x; bi = bse;     }
        if (v.y < best)      { best = v.y; bi = bse + 1; }
        if (v.z < best)      { best = v.z; bi = bse + 2; }
        if (v.w < best)      { best = v.w; bi = bse + 3; }
    }
    // wave reduction, first-occurrence tie break (smaller index wins on equality)
    #pragma unroll
    for (int off = 16; off > 0; off >>= 1) {
        float ov = __shfl_down(best, off, 32);
        int   oi = __shfl_down(bi,   off, 32);
        if (ov < best || (ov == best && oi < bi)) { best = ov; bi = oi; }
    }
    if (lane == 0) idx_out[row] = bi;
}

// ---------------------------------------------------------------------------
// Kernel 1: P2[p][j] = sum_k prototypes[p][k] * W1[1024+k][j]   (512x128, tiny)
// ---------------------------------------------------------------------------
__global__ __launch_bounds__(128) void proto_partial(const float* __restrict__ protos,
                                                     const float* __restrict__ W1,
                                                     float* __restrict__ P2) {
    const int p = blockIdx.x;
    const int j = threadIdx.x;
    const float* __restrict__ pr = protos + (size_t)p * EMBED_DIM;
    const float* __restrict__ wb = W1 + (size_t)EMBED_DIM * HIDDEN;  // bottom half of W1
    float s = 0.f;
    for (int k = 0; k < EMBED_DIM; ++k)
        s += pr[k] * wb[(size_t)k * HIDDEN + j];
    P2[(size_t)p * HIDDEN + j] = s;
}

// ---------------------------------------------------------------------------
// Kernel 2: fused  h = relu(q @ W1_top + P2[idx] + b1);  out = sigmoid(h @ W2 + b2)
// 256 threads = 8 waves. Wave w -> columns [16w,16w+16), 4 M-subtiles of 16 rows.
// fp32 WMMA 16x16x4; A-tile double-buffered via async global->LDS when available.
// ---------------------------------------------------------------------------
#if HAVE_ASYNC_LDS
#define NBUF 2
#else
#define NBUF 1
#endif

__global__ __launch_bounds__(256) void fused_gemm_mlp(const float* __restrict__ q,
                                                      const float* __restrict__ W1,
                                                      const float* __restrict__ b1,
                                                      const float* __restrict__ W2,
                                                      const float* __restrict__ b2,
                                                      const float* __restrict__ P2,
                                                      const int*   __restrict__ idx,
                                                      float* __restrict__ out) {
    __shared__ float As[NBUF][M_TILE * LDS_STRIDE];   // A-tile chunks; As[0] reused for h
    __shared__ int   ids[M_TILE];
    __shared__ float w2s[HIDDEN];

    const int t    = threadIdx.x;
    const int wave = t >> 5;
    const int lane = t & 31;
    const int nm   = lane & 15;                 // M (for A) / N (for B,C,D) within tile
    const int koff = (lane < 16) ? 0 : 2;       // K sub-offset per ISA f32 layouts
    const int col0 = wave * 16;                 // this wave's HIDDEN column base
    const int tile = blockIdx.x * M_TILE;       // first query row of this block

    if (t < M_TILE) ids[t] = idx[tile + t];
    if (t < HIDDEN) w2s[t] = W2[t];

    v8f acc[4] = {};                            // 4 M-subtiles x 16x16 f32
    const int bo = col0 + nm + koff * HIDDEN;   // per-lane W1 offset inside a k-step
    const int NCHUNK = EMBED_DIM / K_CHUNK;     // 8

#if HAVE_ASYNC_LDS
    // each lane copies rows {wave+8i}, 16B at column lane*4 (whole 128-col chunk/block)
    #pragma unroll
    for (int i = 0; i < 8; ++i) {
        const int row = wave + 8 * i;
        async_ld128(q + (size_t)(tile + row) * EMBED_DIM + lane * 4,
                    &As[0][row * LDS_STRIDE + lane * 4]);
    }
    for (int c = 0; c < NCHUNK; ++c) {
        WAIT_ASYNC0();                          // my chunk-c copies landed in LDS
        __syncthreads();                        // everyone's landed; prev compute done
        if (c + 1 < NCHUNK) {                   // prefetch next chunk into other buffer
            const int kc = (c + 1) * K_CHUNK;
            #pragma unroll
            for (int i = 0; i < 8; ++i) {
                const int row = wave + 8 * i;
                async_ld128(q + (size_t)(tile + row) * EMBED_DIM + kc + lane * 4,
                            &As[(c + 1) & 1][row * LDS_STRIDE + lane * 4]);
            }
        }
        const float* Ab = As[c & 1];
        const int kcb = c * K_CHUNK;
        #pragma unroll 4
        for (int s = 0; s < K_CHUNK / 4; ++s) {
            const int k = s * 4;
            v2f b;
            const int bidx = (kcb + k) * HIDDEN + bo;
            b.x = W1[bidx];
            b.y = W1[bidx + HIDDEN];
            #pragma unroll
            for (int sub = 0; sub < 4; ++sub) {
                v2f a = *(const v2f*)(&Ab[(sub * 16 + nm) * LDS_STRIDE + k + koff]);
                acc[sub] = __builtin_amdgcn_wmma_f32_16x16x4_f32(
                    false, a, false, b, (short)0, acc[sub], false, false);
            }
        }
        __syncthreads();                        // all reads of As[c&1] done before reuse
    }
#else
    for (int kc = 0; kc < EMBED_DIM; kc += K_CHUNK) {
        __syncthreads();
        #pragma unroll
        for (int i = 0; i < 8; ++i) {
            int f   = t + 256 * i;
            int row = f >> 5;
            int c4  = f & 31;
            float4 v = *(const float4*)(q + (size_t)(tile + row) * EMBED_DIM + kc + c4 * 4);
            *(float4*)(&As[0][row * LDS_STRIDE + c4 * 4]) = v;
        }
        __syncthreads();
        #pragma unroll 4
        for (int s = 0; s < K_CHUNK / 4; ++s) {
            const int k = s * 4;
            v2f b;
            const int bidx = (kc + k) * HIDDEN + bo;
            b.x = W1[bidx];
            b.y = W1[bidx + HIDDEN];
            #pragma unroll
            for (int sub = 0; sub < 4; ++sub) {
                v2f a = *(const v2f*)(&As[0][(sub * 16 + nm) * LDS_STRIDE + k + koff]);
                acc[sub] = __builtin_amdgcn_wmma_f32_16x16x4_f32(
                    false, a, false, b, (short)0, acc[sub], false, false);
            }
        }
    }
    __syncthreads();
#endif

    // ---- epilogue: += P2[idx] + b1, ReLU, stash h into LDS (reuse As[0]) ----
    const int   j   = col0 + nm;
    const float b1v = b1[j];
    #pragma unroll
    for (int sub = 0; sub < 4; ++sub) {
        #pragma unroll
        for (int r = 0; r < 8; ++r) {
            const int lrow = sub * 16 + ((lane < 16) ? r : 8 + r);  // C/D layout
            float v = acc[sub][r] + b1v + P2[(size_t)ids[lrow] * HIDDEN + j];
            As[0][lrow * LDS_STRIDE + j] = fmaxf(v, 0.f);
        }
    }
    __syncthreads();

    // ---- out = sigmoid(h @ W2 + b2): 64 threads, one row each ----
    if (t < M_TILE) {
        float s = b2[0];
        #pragma unroll 8
        for (int jj = 0; jj < HIDDEN; ++jj)
            s += As[0][t * LDS_STRIDE + jj] * w2s[jj];
        out[tile + t] = 1.f / (1.f + __expf(-s));
    }
}

// ---------------------------------------------------------------------------
extern "C" void kernel_launch(void* const* d_in, const int* in_sizes, int n_in,
                              void* d_out, int out_size, void* d_ws, size_t ws_size,
                              hipStream_t stream) {
    const float* q      = (const float*)d_in[0];
    const float* protos = (const float*)d_in[1];
    const float* dist   = (const float*)d_in[2];
    const float* W1     = (const float*)d_in[3];
    const float* b1     = (const float*)d_in[4];
    const float* W2     = (const float*)d_in[5];
    const float* b2     = (const float*)d_in[6];
    float* out = (float*)d_out;

    int*   idx_ws = (int*)d_ws;                                            // 65536 * 4 B
    float* P2     = (float*)((char*)d_ws + (size_t)N_QUERY * sizeof(int)); // 512*128*4 B

    argmin_rows<<<N_QUERY / 8, 256, 0, stream>>>(dist, idx_ws);
    proto_partial<<<N_PROTO, 128, 0, stream>>>(protos, W1, P2);
    fused_gemm_mlp<<<N_QUERY / M_TILE, 256, 0, stream>>>(q, W1, b1, W2, b2, P2, idx_ws, out);
}